// CombinedCriterion_3264175145043
// MI455X (gfx1250) — compile-verified
//
#include <hip/hip_runtime.h>

// Problem constants (match reference)
#define B_      32
#define N_      1024
#define DD      3
#define NLEV    5
#define EPS     0.1f
#define INV_EPS 10.0f
#define ITERS   20
#define BIGV    1e8f
#define NEGV    (-1e9f)
#define BETA_   0.01f

typedef __attribute__((ext_vector_type(2))) float v2f;
typedef __attribute__((ext_vector_type(8))) float v8f;
typedef __attribute__((ext_vector_type(4))) int   v4i;

// CDNA5 async global->LDS copy path (probe via __has_builtin; sync fallback).
#if defined(__HIP_DEVICE_COMPILE__)
#if __has_builtin(__builtin_amdgcn_global_load_async_to_lds_b128) && \
    __has_builtin(__builtin_amdgcn_s_wait_asynccnt)
#define ASYNC_LDS 1
#endif
#endif

// b128 payload pointers in explicit global (AS1) / LDS (AS3) address spaces,
// matching the builtin's (v4i addrspace(1)*, v4i addrspace(3)*, imm, imm) proto.
typedef __attribute__((address_space(1))) v4i* gv4i_p;
typedef __attribute__((address_space(3))) v4i* lv4i_p;

// Copy N_ float4s (16 KB) global -> LDS cooperatively (256 threads).
__device__ __forceinline__ void stage_cols(const float4* __restrict__ src,
                                           float4* dst, int t)
{
#ifdef ASYNC_LDS
#pragma unroll
  for (int i = t; i < N_; i += 256)
    __builtin_amdgcn_global_load_async_to_lds_b128(
        (gv4i_p)(src + i), (lv4i_p)(dst + i), 0, 0);
#else
#pragma unroll
  for (int i = t; i < N_; i += 256) dst[i] = src[i];
#endif
}

__device__ __forceinline__ void stage_wait()
{
#ifdef ASYNC_LDS
  __builtin_amdgcn_s_wait_asynccnt(0);
#endif
  __syncthreads();
}

// ---------------------------------------------------------------------------
// 16x16 squared-distance tile via V_WMMA_F32_16X16X4_F32 from LDS-staged cols.
//   A (16x4):  lane L holds row (L&15); VGPRs hold K={0,1} (L<16) or {2,3}.
//   D (16x16): VGPR i, lane L -> row i + 8*(L>=16), col (L&15).
// A carries -2*p (hoisted, loop-invariant); c4 = (q0,q1,q2,|q|^2) from LDS.
// ---------------------------------------------------------------------------
__device__ __forceinline__ void sqdist_tile(const v2f A, const float4 c4, bool hi,
                                            const float xr[8], float dist[8])
{
  v2f Bm;
  Bm.x = hi ? c4.z : c4.x;
  Bm.y = hi ? 0.0f : c4.y;
  v8f acc = {};
  acc = __builtin_amdgcn_wmma_f32_16x16x4_f32(false, A, false, Bm,
                                              (short)0, acc, false, false);
#pragma unroll
  for (int i = 0; i < 8; ++i) {
    float v = acc[i] + xr[i] + c4.w;
    dist[i] = v > 0.0f ? v : 0.0f;
  }
}

// Common per-wave tile coordinates: 8 waves/block, 8 blocks/batch.
#define TILE_PROLOGUE                                                        \
  const int  b    = blockIdx.x >> 3;                                         \
  const int  lane = (int)(threadIdx.x & 31);                                 \
  const int  tile = ((blockIdx.x & 7) << 3) | (int)(threadIdx.x >> 5);       \
  const int  n0   = tile * 16;                                               \
  const int  lo   = lane & 15;                                               \
  const bool hi   = lane >= 16;                                              \
  const int  rbase = n0 + ((lane & 16) >> 1);

__device__ __forceinline__ v2f make_A(const float4 pr, bool hi)
{
  v2f A;
  A.x = -2.0f * (hi ? pr.z : pr.x);
  A.y = hi ? 0.0f : (-2.0f * pr.y);
  return A;
}

// ---------------------------------------------------------------------------
// Prep: mask counts, packed (coord,norm) float4s, log_a/log_b, zero f/g/emd.
// ---------------------------------------------------------------------------
__global__ void prep_kernel(const float* __restrict__ x, const unsigned char* __restrict__ xm,
                            const float* __restrict__ y, const unsigned char* __restrict__ ym,
                            float4* __restrict__ packX, float4* __restrict__ packY,
                            float* __restrict__ log_a, float* __restrict__ log_b,
                            float* __restrict__ f, float* __restrict__ g,
                            float* __restrict__ nx, float* __restrict__ ny,
                            float* __restrict__ emd_acc)
{
  const int b = blockIdx.x;
  const int t = threadIdx.x;
  __shared__ float sx[256], sy[256];
  float cx = 0.f, cy = 0.f;
  for (int n = t; n < N_; n += 256) {
    const size_t idx = (size_t)b * N_ + n;
    cx += xm[idx] ? 1.f : 0.f;
    cy += ym[idx] ? 1.f : 0.f;
    const float* xr = x + idx * DD;
    const float* yr = y + idx * DD;
    packX[idx] = make_float4(xr[0], xr[1], xr[2],
                             xr[0]*xr[0] + xr[1]*xr[1] + xr[2]*xr[2]);
    packY[idx] = make_float4(yr[0], yr[1], yr[2],
                             yr[0]*yr[0] + yr[1]*yr[1] + yr[2]*yr[2]);
    f[idx] = 0.f;
    g[idx] = 0.f;
  }
  sx[t] = cx; sy[t] = cy;
  __syncthreads();
  for (int s = 128; s > 0; s >>= 1) {
    if (t < s) { sx[t] += sx[t + s]; sy[t] += sy[t + s]; }
    __syncthreads();
  }
  if (t == 0) { nx[b] = sx[0]; ny[b] = sy[0]; emd_acc[b] = 0.f; }
  const float la = -__logf(sx[0]);
  const float lb = -__logf(sy[0]);
  for (int n = t; n < N_; n += 256) {
    const size_t idx = (size_t)b * N_ + n;
    log_a[idx] = xm[idx] ? la : NEGV;
    log_b[idx] = ym[idx] ? lb : NEGV;
  }
}

// ---------------------------------------------------------------------------
// Chamfer row-min: out[b,n] = min_m (qmask[m] ? d : BIG)  (bias-add form)
// ---------------------------------------------------------------------------
__global__ void __launch_bounds__(256)
rowmin_kernel(const float4* __restrict__ packP, const float4* __restrict__ packQ,
              const unsigned char* __restrict__ qmask, float* __restrict__ out)
{
  TILE_PROLOGUE
  __shared__ float4 scol[N_];
  __shared__ float  sbias[N_];
  stage_cols(packQ + (size_t)b * N_, scol, (int)threadIdx.x);
#pragma unroll
  for (int i = (int)threadIdx.x; i < N_; i += 256)
    sbias[i] = qmask[(size_t)b * N_ + i] ? 0.f : BIGV;
  stage_wait();

  const v2f A = make_A(packP[(size_t)b * N_ + n0 + lo], hi);
  float xr[8];
#pragma unroll
  for (int i = 0; i < 8; ++i) xr[i] = packP[(size_t)b * N_ + rbase + i].w;

  float mn[8];
#pragma unroll
  for (int i = 0; i < 8; ++i) mn[i] = BIGV;

  for (int m0 = 0; m0 < N_; m0 += 16) {
    float dist[8];
    sqdist_tile(A, scol[m0 + lo], hi, xr, dist);
    const float bias = sbias[m0 + lo];
#pragma unroll
    for (int i = 0; i < 8; ++i) mn[i] = fminf(mn[i], dist[i] + bias);
  }
#pragma unroll
  for (int off = 1; off < 16; off <<= 1) {
#pragma unroll
    for (int i = 0; i < 8; ++i)
      mn[i] = fminf(mn[i], __shfl_xor(mn[i], off, 32));
  }
  if ((lane & 15) == 0) {
#pragma unroll
    for (int i = 0; i < 8; ++i)
      out[(size_t)b * N_ + rbase + i] = mn[i];
  }
}

// ---------------------------------------------------------------------------
// Sinkhorn half-iteration (online log-sum-exp, branchless):
//   hout[n] = EPS * ( logw[n] - LSE_m( (gEff[m] - C[n,m]) / EPS ) )
// Column mask folded into staged gEff (= -BIG when masked -> exp underflows
// to exactly 0, same as reference's C=BIG). Row mask handled per element.
// ---------------------------------------------------------------------------
__global__ void __launch_bounds__(256)
sinkhorn_lse_kernel(const float4* __restrict__ packP, const float4* __restrict__ packQ,
                    const unsigned char* __restrict__ pmask,
                    const unsigned char* __restrict__ qmask,
                    const float* __restrict__ logw,
                    const float* __restrict__ hin,
                    float* __restrict__ hout)
{
  TILE_PROLOGUE
  __shared__ float4 scol[N_];
  __shared__ float  sg[N_];
  stage_cols(packQ + (size_t)b * N_, scol, (int)threadIdx.x);
#pragma unroll
  for (int i = (int)threadIdx.x; i < N_; i += 256) {
    const size_t idx = (size_t)b * N_ + i;
    sg[i] = qmask[idx] ? hin[idx] : -BIGV;
  }
  stage_wait();

  const v2f A = make_A(packP[(size_t)b * N_ + n0 + lo], hi);
  float xr[8];
  bool  rm[8];
#pragma unroll
  for (int i = 0; i < 8; ++i) {
    xr[i] = packP[(size_t)b * N_ + rbase + i].w;
    rm[i] = pmask[(size_t)b * N_ + rbase + i] != 0;
  }
  float mx[8], sm[8];
#pragma unroll
  for (int i = 0; i < 8; ++i) { mx[i] = -__builtin_inff(); sm[i] = 0.f; }

  for (int m0 = 0; m0 < N_; m0 += 16) {
    float dist[8];
    sqdist_tile(A, scol[m0 + lo], hi, xr, dist);
    const float gc = sg[m0 + lo];
#pragma unroll
    for (int i = 0; i < 8; ++i) {
      const float c  = rm[i] ? dist[i] : BIGV;
      const float z  = (gc - c) * INV_EPS;
      const float nm = fmaxf(mx[i], z);
      sm[i] = sm[i] * __expf(mx[i] - nm) + __expf(z - nm);
      mx[i] = nm;
    }
  }
  // combine (max, sumexp) across the 16 lanes sharing each row group
#pragma unroll
  for (int off = 1; off < 16; off <<= 1) {
#pragma unroll
    for (int i = 0; i < 8; ++i) {
      const float om = __shfl_xor(mx[i], off, 32);
      const float os = __shfl_xor(sm[i], off, 32);
      const float nm = fmaxf(mx[i], om);
      sm[i] = sm[i] * __expf(mx[i] - nm) + os * __expf(om - nm);
      mx[i] = nm;
    }
  }
  if ((lane & 15) == 0) {
#pragma unroll
    for (int i = 0; i < 8; ++i) {
      const size_t idx = (size_t)b * N_ + rbase + i;
      hout[idx] = EPS * (logw[idx] - (mx[i] + __logf(sm[i])));
    }
  }
}

// ---------------------------------------------------------------------------
// EMD:  emd_acc[b] += sum_{n,m} exp((f+g-C)/EPS) * C
// Column mask folded into staged gEff (= -2*BIG -> exp == 0 exactly).
// ---------------------------------------------------------------------------
__global__ void __launch_bounds__(256)
emd_kernel(const float4* __restrict__ packP, const float4* __restrict__ packQ,
           const unsigned char* __restrict__ pmask,
           const unsigned char* __restrict__ qmask,
           const float* __restrict__ f, const float* __restrict__ g,
           float* __restrict__ emd_acc)
{
  TILE_PROLOGUE
  __shared__ float4 scol[N_];
  __shared__ float  sg[N_];
  stage_cols(packQ + (size_t)b * N_, scol, (int)threadIdx.x);
#pragma unroll
  for (int i = (int)threadIdx.x; i < N_; i += 256) {
    const size_t idx = (size_t)b * N_ + i;
    sg[i] = qmask[idx] ? g[idx] : -2.0f * BIGV;
  }
  stage_wait();

  const v2f A = make_A(packP[(size_t)b * N_ + n0 + lo], hi);
  float xr[8], fr[8];
  bool  rm[8];
#pragma unroll
  for (int i = 0; i < 8; ++i) {
    const size_t idx = (size_t)b * N_ + rbase + i;
    xr[i] = packP[idx].w;
    fr[i] = f[idx];
    rm[i] = pmask[idx] != 0;
  }
  float local = 0.f;
  for (int m0 = 0; m0 < N_; m0 += 16) {
    float dist[8];
    sqdist_tile(A, scol[m0 + lo], hi, xr, dist);
    const float gc = sg[m0 + lo];
#pragma unroll
    for (int i = 0; i < 8; ++i) {
      const float c = rm[i] ? dist[i] : BIGV;
      local += __expf((fr[i] + gc - c) * INV_EPS) * c;  // exp underflows to 0 when masked
    }
  }
#pragma unroll
  for (int off = 1; off < 32; off <<= 1)
    local += __shfl_xor(local, off, 32);
  if (lane == 0) atomicAdd(&emd_acc[b], local);
}

// ---------------------------------------------------------------------------
// Chamfer per-batch reduction: cd_acc[b] = sum(xm?dxy)/nx + sum(ym?dyx)/ny
// ---------------------------------------------------------------------------
__global__ void chamfer_reduce_kernel(const float* __restrict__ dxy, const float* __restrict__ dyx,
                                      const unsigned char* __restrict__ xm,
                                      const unsigned char* __restrict__ ym,
                                      const float* __restrict__ nx, const float* __restrict__ ny,
                                      float* __restrict__ cd_acc)
{
  const int b = blockIdx.x;
  const int t = threadIdx.x;
  __shared__ float s1[256], s2[256];
  float a = 0.f, c = 0.f;
  for (int n = t; n < N_; n += 256) {
    const size_t idx = (size_t)b * N_ + n;
    a += xm[idx] ? dxy[idx] : 0.f;
    c += ym[idx] ? dyx[idx] : 0.f;
  }
  s1[t] = a; s2[t] = c;
  __syncthreads();
  for (int s = 128; s > 0; s >>= 1) {
    if (t < s) { s1[t] += s1[t + s]; s2[t] += s2[t + s]; }
    __syncthreads();
  }
  if (t == 0) cd_acc[b] = s1[0] / nx[b] + s2[0] / ny[b];
}

// ---------------------------------------------------------------------------
// Final: loss, kl_loss, recon_loss, topdown_kl[5]  -> out[0..7]
// ---------------------------------------------------------------------------
__global__ void final_kernel(const float* __restrict__ kls,
                             const float* __restrict__ cd_acc,
                             const float* __restrict__ emd_acc,
                             float* __restrict__ out)
{
  const int t = threadIdx.x;  // single wave of 32
  float klsum = 0.f, cds = 0.f, emds = 0.f;
  for (int i = t; i < B_ * NLEV; i += 32) klsum += kls[i];
  for (int i = t; i < B_; i += 32) { cds += cd_acc[i]; emds += emd_acc[i]; }
#pragma unroll
  for (int off = 1; off < 32; off <<= 1) {
    klsum += __shfl_xor(klsum, off, 32);
    cds   += __shfl_xor(cds,   off, 32);
    emds  += __shfl_xor(emds,  off, 32);
  }
  if (t == 0) {
    const float kl_loss = klsum / (float)B_;
    const float recon   = (cds + emds) / (float)B_;
    out[0] = BETA_ * kl_loss + recon;
    out[1] = kl_loss;
    out[2] = recon;
  }
  if (t < NLEV) {
    float s = 0.f;
    for (int bb = 0; bb < B_; ++bb) s += kls[bb * NLEV + t];
    const float zs[NLEV] = {2.f, 4.f, 8.f, 16.f, 32.f};
    out[3 + t] = (s / (float)B_) / (zs[t] * 16.0f);
  }
}

// ---------------------------------------------------------------------------
extern "C" void kernel_launch(void* const* d_in, const int* in_sizes, int n_in,
                              void* d_out, int out_size, void* d_ws, size_t ws_size,
                              hipStream_t stream)
{
  (void)in_sizes; (void)n_in; (void)out_size; (void)ws_size;
  const float*         x   = (const float*)d_in[0];
  const unsigned char* xm  = (const unsigned char*)d_in[1];  // jnp bool = 1 byte
  const float*         y   = (const float*)d_in[2];
  const unsigned char* ym  = (const unsigned char*)d_in[3];
  const float*         kls = (const float*)d_in[4];
  float* out = (float*)d_out;

  // workspace carve-up (~1.8 MB); float4 arrays first for 16B alignment
  float* w = (float*)d_ws;
  const size_t BN = (size_t)B_ * N_;
  float4* packX = (float4*)w; w += 4 * BN;
  float4* packY = (float4*)w; w += 4 * BN;
  float* log_a  = w; w += BN;
  float* log_b  = w; w += BN;
  float* f      = w; w += BN;
  float* g      = w; w += BN;
  float* dxy    = w; w += BN;
  float* dyx    = w; w += BN;
  float* nx     = w; w += B_;
  float* ny     = w; w += B_;
  float* cd_acc = w; w += B_;
  float* emd    = w; w += B_;

  prep_kernel<<<B_, 256, 0, stream>>>(x, xm, y, ym, packX, packY, log_a, log_b,
                                      f, g, nx, ny, emd);

  const int tile_blocks = B_ * 8;  // 8 wave32 tiles per 256-thread block

  // Chamfer (both directions; d is symmetric so swap p/q roles)
  rowmin_kernel<<<tile_blocks, 256, 0, stream>>>(packX, packY, ym, dxy);
  rowmin_kernel<<<tile_blocks, 256, 0, stream>>>(packY, packX, xm, dyx);
  chamfer_reduce_kernel<<<B_, 256, 0, stream>>>(dxy, dyx, xm, ym, nx, ny, cd_acc);

  // Sinkhorn: f from g, then g from f (g initialized to 0 in prep)
  for (int it = 0; it < ITERS; ++it) {
    sinkhorn_lse_kernel<<<tile_blocks, 256, 0, stream>>>(packX, packY, xm, ym,
                                                         log_a, g, f);
    sinkhorn_lse_kernel<<<tile_blocks, 256, 0, stream>>>(packY, packX, ym, xm,
                                                         log_b, f, g);
  }

  emd_kernel<<<tile_blocks, 256, 0, stream>>>(packX, packY, xm, ym, f, g, emd);
  final_kernel<<<1, 32, 0, stream>>>(kls, cd_acc, emd, out);
}